// GCONV_61091614818591
// MI455X (gfx1250) — compile-verified
//
#include <hip/hip_runtime.h>
#include <hip/hip_bf16.h>

// ---------------------------------------------------------------------------
// DCRNN diffusion graph conv for MI455X (gfx1250, wave32).
//  Phase 1: transpose inputs [B, N*D] -> x0 [N, D, B]          (LDS tiles)
//  Phase 2: 4x scatter-add SPMM with global_atomic_add_f32;
//           Chebyshev x2 = 2*A*x1 - x0 folds -x0 into the init.
//  Phase 3: [B*N,330] @ [330,64] + bias with V_WMMA_F32_16X16X4_F32.
//           A-tiles + weight slice staged in LDS in a pair-packed layout so
//           the inner loop is branch-free: ds_load_b64 x2 + v_wmma per step.
// Everything fp32 (memory-bound; 23.3 TB/s HBM dominates, not FLOPs).
// Requires ws_size >= 5 * N*D*B * 4 bytes (~845 MB).
// ---------------------------------------------------------------------------

namespace {
constexpr int Bb  = 64;          // batch
constexpr int Nn  = 10000;       // nodes
constexpr int Dd  = 66;          // feature dim
constexpr int Oo  = 64;          // output dim
constexpr int Mm  = 5;           // 2*K+1 Chebyshev matrices
constexpr int KK  = Dd * Mm;     // 330 contraction length
constexpr int KP  = 332;         // padded to multiple of 4
constexpr int NG  = KP / 2;      // 166 pair-groups
constexpr int TIL = NG * 32;     // 5312 floats per packed 16-row tile
constexpr int DB  = Dd * Bb;     // 4224 row width of x buffers
constexpr int ND  = Nn * Dd;     // 660000
constexpr int NDB = Nn * DB;     // 42,240,000 floats per buffer
constexpr int NO  = Nn * Oo;     // 640,000
constexpr int F4  = DB / 4;      // 1056 float4 per row
}

typedef float v2f __attribute__((ext_vector_type(2)));
typedef float v8f __attribute__((ext_vector_type(8)));

// ---- Phase 1: [B, ND] -> [ND, B] transpose via LDS 64x64 tiles -------------
__global__ void gconv_transpose(const float* __restrict__ in,
                                float* __restrict__ x0) {
    __shared__ float tile[64][65];
    const int nd0 = blockIdx.x * 64;
    const int tx  = threadIdx.x;   // 0..63
    const int ty  = threadIdx.y;   // 0..3
    if (nd0 + tx < ND) {
        for (int b = ty; b < 64; b += 4)
            tile[tx][b] = in[b * ND + nd0 + tx];   // coalesced along nd
    }
    __syncthreads();
    for (int i = ty; i < 64; i += 4) {
        if (nd0 + i < ND)
            x0[(nd0 + i) * Bb + tx] = tile[i][tx]; // coalesced along b
    }
}

// ---- init kernels ----------------------------------------------------------
__global__ void gconv_zero4(float4* __restrict__ y, int n4) {
    int i = blockIdx.x * blockDim.x + threadIdx.x;
    if (i < n4) y[i] = make_float4(0.f, 0.f, 0.f, 0.f);
}

__global__ void gconv_neg4(float4* __restrict__ y,
                           const float4* __restrict__ x, int n4) {
    int i = blockIdx.x * blockDim.x + threadIdx.x;
    if (i < n4) {
        float4 v = x[i];
        y[i] = make_float4(-v.x, -v.y, -v.z, -v.w);
    }
}

// ---- Phase 2: SPMM scatter-add: y[r, :] += scale*v * x[c, :] ---------------
__global__ void gconv_spmm(const int* __restrict__ rows,
                           const int* __restrict__ cols,
                           const float* __restrict__ vals,
                           const float* __restrict__ x,
                           float* __restrict__ y,
                           float scale, int total) {
    int i = blockIdx.x * blockDim.x + threadIdx.x;
    if (i >= total) return;
    unsigned ui = (unsigned)i;
    int e = (int)(ui / (unsigned)F4);      // edge
    int f = (int)(ui - (unsigned)e * F4);  // float4 chunk within row
    int r = rows[e];
    int c = cols[e];
    float v = vals[e] * scale;
    float4 xv = ((const float4*)(x + (size_t)c * DB))[f];
    float* yp = y + (size_t)r * DB + (size_t)f * 4;
    unsafeAtomicAdd(yp + 0, v * xv.x);     // -> global_atomic_add_f32
    unsafeAtomicAdd(yp + 1, v * xv.y);
    unsafeAtomicAdd(yp + 2, v * xv.z);
    unsafeAtomicAdd(yp + 3, v * xv.w);
}

// ---- Phase 3: WMMA fp32 GEMM -----------------------------------------------
// Contraction order permuted to k' = m*66 + d (order-independent sum) so the
// staging loops have compile-time-fixed source pointers. LDS pair layout:
//   buf[(k'>>1)*32 + col*2 + (k'&1)]  ->  one ds_load_b64 per fragment pair.
__global__ void __launch_bounds__(128)
gconv_gemm(const float* __restrict__ x0,  const float* __restrict__ x10,
           const float* __restrict__ x20, const float* __restrict__ x11,
           const float* __restrict__ x21, const float* __restrict__ W,
           const float* __restrict__ bias, float* __restrict__ out) {
    __shared__ float As[4 * TIL];   // per-wave A tiles (16 b-rows x 332 k')
    __shared__ float Ws[TIL];       // shared weight slice (332 k' x 16 o)

    const int tid  = threadIdx.x;
    const int wave = tid >> 5;
    const int lane = tid & 31;
    const int n    = blockIdx.x * 4 + wave;    // one n per wave
    const int b0   = blockIdx.y * 16;          // 16 batch rows per tile
    const int o0   = blockIdx.z * 16;          // 16 output cols per tile

    // Stage weight slice (block-wide), permuted + pair-packed + zero-padded.
    for (int idx = tid; idx < KP * 16; idx += 128) {
        int kp = idx >> 4;
        int c2 = idx & 15;
        float v = 0.0f;
        if (kp < KK) {
            int m = (unsigned)kp / 66u;
            int d = kp - m * 66;
            v = W[(d * Mm + m) * Oo + o0 + c2];
        }
        Ws[(kp >> 1) * 32 + c2 * 2 + (kp & 1)] = v;
    }

    // Stage this wave's A tile: fixed pointer per m (no dynamic selection).
    float* Aw = As + wave * TIL;
    const size_t nbase = (size_t)n * DB + b0;
    const float* srcs[1]; (void)srcs;
    {
        const float* p;
        int m;
        #define GCONV_STAGE(PTR, MM)                                        \
            p = (PTR); m = (MM);                                            \
            for (int idx = lane; idx < Dd * 16; idx += 32) {                \
                int d = idx >> 4;                                           \
                int b = idx & 15;                                           \
                int kp = m * Dd + d;                                        \
                Aw[(kp >> 1) * 32 + b * 2 + (kp & 1)] = p[nbase + d * Bb + b]; \
            }
        GCONV_STAGE(x0, 0)
        GCONV_STAGE(x10, 1)
        GCONV_STAGE(x20, 2)
        GCONV_STAGE(x11, 3)
        GCONV_STAGE(x21, 4)
        #undef GCONV_STAGE
    }
    // Zero-pad group 165 (k' = 330, 331): 32 floats, one per lane.
    Aw[165 * 32 + lane] = 0.0f;

    __syncthreads();

    // Branch-free WMMA loop: per step, A pair + B pair from LDS (b64 loads).
    const int col = lane & 15;
    const int h1  = lane >> 4;                 // upper half-wave: K+2, K+3
    const float* Ap = Aw + col * 2 + h1 * 32;  // 8-byte aligned
    const float* Bp = Ws + col * 2 + h1 * 32;

    v8f c = {};
    #pragma unroll
    for (int kk = 0; kk < KP / 4; ++kk) {
        v2f a = *(const v2f*)(Ap + kk * 64);
        v2f b = *(const v2f*)(Bp + kk * 64);
        c = __builtin_amdgcn_wmma_f32_16x16x4_f32(
                false, a, false, b, (short)0, c, false, false);
    }

    const float bv      = bias[o0 + col];
    const int   rowbase = b0 + (h1 << 3);      // M = 0..7 or 8..15
    #pragma unroll
    for (int i = 0; i < 8; ++i) {
        out[(size_t)(rowbase + i) * NO + (size_t)n * Oo + o0 + col] = c[i] + bv;
    }
}

// ---------------------------------------------------------------------------
extern "C" void kernel_launch(void* const* d_in, const int* in_sizes, int n_in,
                              void* d_out, int out_size, void* d_ws, size_t ws_size,
                              hipStream_t stream) {
    const float* inputs = (const float*)d_in[0];
    // d_in[1] = state (unused by forward)
    const float* weight = (const float*)d_in[2];
    const float* biases = (const float*)d_in[3];
    const float* vals0  = (const float*)d_in[4];
    const float* vals1  = (const float*)d_in[5];
    const int*   rows0  = (const int*)d_in[6];
    const int*   cols0  = (const int*)d_in[7];
    const int*   rows1  = (const int*)d_in[8];
    const int*   cols1  = (const int*)d_in[9];
    float*       out    = (float*)d_out;

    const int E = in_sizes[4];

    float* x0  = (float*)d_ws;
    float* x10 = x0  + (size_t)NDB;
    float* x20 = x10 + (size_t)NDB;
    float* x11 = x20 + (size_t)NDB;
    float* x21 = x11 + (size_t)NDB;

    // Phase 1: x0 = inputs^T  ([B, N*D] -> [N*D, B])
    gconv_transpose<<<(ND + 63) / 64, dim3(64, 4), 0, stream>>>(inputs, x0);

    const int n4 = NDB / 4;
    const int zg = (n4 + 255) / 256;
    const long long total = (long long)E * F4;
    const int sg = (int)((total + 255) / 256);

    // Phase 2: support 0
    gconv_zero4<<<zg, 256, 0, stream>>>((float4*)x10, n4);
    gconv_spmm <<<sg, 256, 0, stream>>>(rows0, cols0, vals0, x0,  x10, 1.0f, (int)total);
    gconv_neg4 <<<zg, 256, 0, stream>>>((float4*)x20, (const float4*)x0, n4);
    gconv_spmm <<<sg, 256, 0, stream>>>(rows0, cols0, vals0, x10, x20, 2.0f, (int)total);
    // Phase 2: support 1
    gconv_zero4<<<zg, 256, 0, stream>>>((float4*)x11, n4);
    gconv_spmm <<<sg, 256, 0, stream>>>(rows1, cols1, vals1, x0,  x11, 1.0f, (int)total);
    gconv_neg4 <<<zg, 256, 0, stream>>>((float4*)x21, (const float4*)x0, n4);
    gconv_spmm <<<sg, 256, 0, stream>>>(rows1, cols1, vals1, x11, x21, 2.0f, (int)total);

    // Phase 3: WMMA GEMM + bias -> out [B, N*O]
    gconv_gemm<<<dim3(Nn / 4, Bb / 16, Oo / 16), 128, 0, stream>>>(
        x0, x10, x20, x11, x21, weight, biases, out);
}